// ADMM_49641232007393
// MI455X (gfx1250) — compile-verified
//
#include <hip/hip_runtime.h>

// ---------------------------------------------------------------------------
// ADMM sparse coding on MI455X (gfx1250, wave32, WMMA).
// fp32 throughout (matches reference); GEMMs via V_WMMA_F32_16X16X4_F32.
// ---------------------------------------------------------------------------

typedef float v2f __attribute__((ext_vector_type(2)));
typedef float v8f __attribute__((ext_vector_type(8)));

#define ADMM_MU      5e-05f
#define ADMM_LAMBDA  12.5f
#define ADMM_EPS     0.001f
#define ADMM_RHO     0.01f
#define ADMM_BETA    (ADMM_RHO / (2.0f * ADMM_LAMBDA))
#define ADMM_ITERS   100

#define GBM 128
#define GBN 128
#define GBK 16
// 8 waves per block (256 thr, wave32): 2x4 wave grid; each wave -> 64x32 out
// region = 4x2 grid of 16x16 WMMA tiles (v8f accumulators).

// ---------------------------------------------------------------------------
// Generic fp32 WMMA GEMM.
//   TRANS_A=0:  C[m][n] = sum_k A[m][k] * B[k][n]
//   TRANS_A=1:  C[m][n] = sum_k A[k][m] * B[k][n]   (A^T B, e.g. H^T H)
//   MODE=0: store C (+ diag_add on the diagonal, for G = H^T H + rho*I)
//   MODE=1: don't store; accumulate sum((C - X)^2) into err_part[block]
// done_flag: optional device guard (skip all work once ADMM converged).
// ---------------------------------------------------------------------------
template <int TRANS_A, int MODE>
__global__ __launch_bounds__(256) void gemm_f32_wmma(
    const float* __restrict__ A, const float* __restrict__ B,
    float* __restrict__ C, const float* __restrict__ X,
    int M, int N, int K, int lda, int ldb, int ldc,
    float diag_add, float* __restrict__ err_part,
    const int* __restrict__ done_flag) {
  if (done_flag != nullptr && *done_flag != 0) return;

  // TRANS_A=0: As = [128][20] (pad 20 -> conflict-free column reads)
  // TRANS_A=1: As = [16][128] (K-major)
  __shared__ float As[GBM * 20];
  __shared__ float Bs[GBK * GBN];  // [16][128], K-major
  __shared__ float red[256];

  const int tid   = threadIdx.x;
  const int lane  = tid & 31;
  const int wave  = tid >> 5;
  const int waveM = wave >> 2;  // 0..1
  const int waveN = wave & 3;   // 0..3
  const int half  = lane >> 4;  // 0/1
  const int l16   = lane & 15;

  const int m0 = blockIdx.y * GBM;
  const int n0 = blockIdx.x * GBN;

  v8f acc[4][2];
#pragma unroll
  for (int i = 0; i < 4; i++)
#pragma unroll
    for (int j = 0; j < 2; j++)
#pragma unroll
      for (int e = 0; e < 8; e++) acc[i][j][e] = 0.0f;

  for (int kb = 0; kb < K; kb += GBK) {
    // ---- stage A tile ----
    if (TRANS_A == 0) {
      // A[m0+row][kb..kb+15], 128x16, float4 loads, scalar stores (pad 20)
#pragma unroll
      for (int s = 0; s < 2; s++) {
        int slot = tid + s * 256;      // 0..511
        int row  = slot >> 2;          // 0..127
        int c4   = (slot & 3) << 2;    // 0,4,8,12
        const float4 va =
            *(const float4*)&A[(size_t)(m0 + row) * lda + kb + c4];
        As[row * 20 + c4 + 0] = va.x;
        As[row * 20 + c4 + 1] = va.y;
        As[row * 20 + c4 + 2] = va.z;
        As[row * 20 + c4 + 3] = va.w;
      }
    } else {
      // A[kb..kb+15][m0..m0+127], 16x128, K-major in LDS
#pragma unroll
      for (int s = 0; s < 2; s++) {
        int slot = tid + s * 256;
        int row  = slot >> 5;          // 0..15
        int c4   = (slot & 31) << 2;   // 0..124
        *(float4*)&As[row * 128 + c4] =
            *(const float4*)&A[(size_t)(kb + row) * lda + m0 + c4];
      }
    }
    // ---- stage B tile: B[kb..kb+15][n0..n0+127] ----
#pragma unroll
    for (int s = 0; s < 2; s++) {
      int slot = tid + s * 256;
      int row  = slot >> 5;
      int c4   = (slot & 31) << 2;
      *(float4*)&Bs[row * 128 + c4] =
          *(const float4*)&B[(size_t)(kb + row) * ldb + n0 + c4];
    }
    __syncthreads();

    // ---- 4 WMMA k-steps of 4 per LDS tile ----
#pragma unroll
    for (int kk = 0; kk < GBK; kk += 4) {
      // A fragments: lane layout per ISA 7.12.2 (K = vgpr + 2*half, M = l16)
      v2f af[4];
#pragma unroll
      for (int i = 0; i < 4; i++) {
        int row = waveM * 64 + i * 16 + l16;
        if (TRANS_A == 0) {
          af[i].x = As[row * 20 + kk + 2 * half + 0];
          af[i].y = As[row * 20 + kk + 2 * half + 1];
        } else {
          af[i].x = As[(kk + 2 * half + 0) * 128 + row];
          af[i].y = As[(kk + 2 * half + 1) * 128 + row];
        }
      }
      v2f bf[2];
#pragma unroll
      for (int j = 0; j < 2; j++) {
        int col = waveN * 32 + j * 16 + l16;
        bf[j].x = Bs[(kk + 2 * half + 0) * 128 + col];
        bf[j].y = Bs[(kk + 2 * half + 1) * 128 + col];
      }
#pragma unroll
      for (int i = 0; i < 4; i++)
#pragma unroll
        for (int j = 0; j < 2; j++)
          acc[i][j] = __builtin_amdgcn_wmma_f32_16x16x4_f32(
              false, af[i], false, bf[j], (short)0, acc[i][j], false, false);
    }
    __syncthreads();
  }

  // ---- epilogue ----
  if (MODE == 0) {
#pragma unroll
    for (int i = 0; i < 4; i++)
#pragma unroll
      for (int j = 0; j < 2; j++)
#pragma unroll
        for (int r = 0; r < 8; r++) {
          int grow = m0 + waveM * 64 + i * 16 + r + 8 * half;
          int gcol = n0 + waveN * 32 + j * 16 + l16;
          float val = acc[i][j][r];
          if (diag_add != 0.0f && grow == gcol) val += diag_add;
          C[(size_t)grow * ldc + gcol] = val;
        }
  } else {
    float part = 0.0f;
#pragma unroll
    for (int i = 0; i < 4; i++)
#pragma unroll
      for (int j = 0; j < 2; j++)
#pragma unroll
        for (int r = 0; r < 8; r++) {
          int grow = m0 + waveM * 64 + i * 16 + r + 8 * half;
          int gcol = n0 + waveN * 32 + j * 16 + l16;
          float d  = acc[i][j][r] - X[(size_t)grow * ldc + gcol];
          part += d * d;
        }
    red[tid] = part;
    __syncthreads();
    for (int o = 128; o > 0; o >>= 1) {
      if (tid < o) red[tid] += red[tid + o];
      __syncthreads();
    }
    if (tid == 0)
      err_part[blockIdx.y * gridDim.x + blockIdx.x] = red[0];  // deterministic
  }
}

// ---------------------------------------------------------------------------
// Gauss-Jordan in-place inversion (pivot-free; G is SPD: H^T H + rho*I).
// prep: snapshot scaled pivot row + pivot column (no writes to A -> race-free)
// update: A[k][j] = pivrow[j]; else A[i][j] = (j==k?0:A[i][j]) - pivcol[i]*pivrow[j]
// ---------------------------------------------------------------------------
__global__ void gj_prep(const float* __restrict__ A, float* __restrict__ pivrow,
                        float* __restrict__ pivcol, int k, int n) {
  int j = blockIdx.x * blockDim.x + threadIdx.x;
  if (j >= n) return;
  float p = A[(size_t)k * n + k];
  pivrow[j] = (j == k) ? (1.0f / p) : A[(size_t)k * n + j] / p;
  pivcol[j] = A[(size_t)j * n + k];
}

__global__ void gj_update(float* __restrict__ A,
                          const float* __restrict__ pivrow,
                          const float* __restrict__ pivcol, int k, int n) {
  int j = blockIdx.x * blockDim.x + threadIdx.x;
  int i = blockIdx.y;
  float pr = pivrow[j];
  if (i == k) {
    A[(size_t)i * n + j] = pr;
  } else {
    float val = (j == k) ? 0.0f : A[(size_t)i * n + j];
    A[(size_t)i * n + j] = val - pivcol[i] * pr;
  }
}

// ---------------------------------------------------------------------------
// Elementwise ADMM kernels
// ---------------------------------------------------------------------------
__global__ void k_init(float* s, float* u, float* v, int* done, int n) {
  int i = blockIdx.x * blockDim.x + threadIdx.x;
  if (i < n) {
    s[i] = 0.0f;
    u[i] = 0.0f;
    v[i] = 0.0f;
  }
  if (i == 0) *done = 0;
}

__global__ void k_right(const float* __restrict__ Htx,
                        const float* __restrict__ v,
                        const float* __restrict__ u, float* __restrict__ right,
                        int n, const int* __restrict__ done) {
  if (*done) return;
  int i = blockIdx.x * blockDim.x + threadIdx.x;
  if (i < n) right[i] = Htx[i] + ADMM_RHO * (v[i] - u[i]);
}

__global__ __launch_bounds__(256) void k_update(
    float* __restrict__ s, const float* __restrict__ s_new,
    float* __restrict__ u, float* __restrict__ v,
    float* __restrict__ diff_part, int n, const int* __restrict__ done) {
  if (*done) return;
  __shared__ float red[256];
  int tid = threadIdx.x;
  int i = blockIdx.x * 256 + tid;
  float local = 0.0f;
  if (i < n) {
    float sn = s_new[i];
    float so = s[i];
    float uo = u[i];
    local = fabsf(sn - so);
    float z  = sn + uo;
    float vn = copysignf(fmaxf(fabsf(z) - ADMM_BETA, 0.0f), z);  // shrinkage
    u[i] = uo + ADMM_MU * (sn - vn);
    v[i] = vn;
    s[i] = sn;
  }
  red[tid] = local;
  __syncthreads();
  for (int o = 128; o > 0; o >>= 1) {
    if (tid < o) red[tid] += red[tid + o];
    __syncthreads();
  }
  if (tid == 0) diff_part[blockIdx.x] = red[0];  // deterministic partials
}

// One small block: deterministic reduction of convergence sum + error,
// records errors[it] (NaN once done|conv, matching reference), updates done.
__global__ __launch_bounds__(256) void k_flag(
    const float* __restrict__ diff_part, int ndiff,
    const float* __restrict__ err_part, int nerr, int* __restrict__ done,
    float* __restrict__ errors, int it) {
  __shared__ float red[256];
  int tid = threadIdx.x;
  float acc = 0.0f;
  for (int j = tid; j < ndiff; j += 256) acc += diff_part[j];
  red[tid] = acc;
  __syncthreads();
  for (int o = 128; o > 0; o >>= 1) {
    if (tid < o) red[tid] += red[tid + o];
    __syncthreads();
  }
  if (tid == 0) {
    float diff = red[0];
    float err = 0.0f;
    for (int j = 0; j < nerr; j++) err += err_part[j];
    int d = *done;
    int conv = (diff <= ADMM_EPS) ? 1 : 0;
    errors[it] = (d || conv) ? __uint_as_float(0x7FC00000u) : err;
    *done = d | conv;
  }
}

// ---------------------------------------------------------------------------
// Host orchestration (graph-capturable: only kernel launches on `stream`).
// ---------------------------------------------------------------------------
extern "C" void kernel_launch(void* const* d_in, const int* in_sizes, int n_in,
                              void* d_out, int out_size, void* d_ws,
                              size_t ws_size, hipStream_t stream) {
  (void)in_sizes; (void)n_in; (void)out_size; (void)ws_size;
  const int N = 1024, M = 2048, Bc = 512;

  const float* H = (const float*)d_in[0];  // [N, M]
  const float* x = (const float*)d_in[1];  // [N, B]

  float* s_out  = (float*)d_out;                 // [M*B]
  float* errors = s_out + (size_t)M * Bc;        // [100]

  char* ws = (char*)d_ws;
  size_t off = 0;
  auto walloc = [&](size_t bytes) -> void* {
    void* p = ws + off;
    off += (bytes + 255) & ~(size_t)255;
    return p;
  };
  float* G      = (float*)walloc((size_t)M * M * 4);   // becomes `left`
  float* Htx    = (float*)walloc((size_t)M * Bc * 4);
  float* right  = (float*)walloc((size_t)M * Bc * 4);
  float* s_new  = (float*)walloc((size_t)M * Bc * 4);
  float* u      = (float*)walloc((size_t)M * Bc * 4);
  float* v      = (float*)walloc((size_t)M * Bc * 4);
  float* pivrow = (float*)walloc((size_t)M * 4);
  float* pivcol = (float*)walloc((size_t)M * 4);
  const int nel = M * Bc;
  const int ndiff = (nel + 255) / 256;                 // 4096 blocks
  float* diff_p = (float*)walloc((size_t)ndiff * 4);
  const int nerr = (Bc / GBN) * (N / GBM);             // 32 blocks
  float* err_p  = (float*)walloc((size_t)nerr * 4);
  int* done     = (int*)walloc(256);

  // init state (s lives in d_out; zero everything each call)
  k_init<<<ndiff, 256, 0, stream>>>(s_out, u, v, done, nel);

  // G = H^T H + rho*I   (M x M, K = N)
  gemm_f32_wmma<1, 0><<<dim3(M / GBN, M / GBM), 256, 0, stream>>>(
      H, H, G, nullptr, M, M, N, M, M, M, ADMM_RHO, nullptr, nullptr);
  // Htx = H^T x         (M x B, K = N)
  gemm_f32_wmma<1, 0><<<dim3(Bc / GBN, M / GBM), 256, 0, stream>>>(
      H, x, Htx, nullptr, M, Bc, N, M, Bc, Bc, 0.0f, nullptr, nullptr);

  // left = inv(G) via pivot-free Gauss-Jordan (SPD)
  for (int k = 0; k < M; k++) {
    gj_prep<<<M / 256, 256, 0, stream>>>(G, pivrow, pivcol, k, M);
    gj_update<<<dim3(M / 256, M), 256, 0, stream>>>(G, pivrow, pivcol, k, M);
  }

  // ADMM iterations (state frozen on-device once converged)
  for (int it = 0; it < ADMM_ITERS; it++) {
    k_right<<<ndiff, 256, 0, stream>>>(Htx, v, u, right, nel, done);
    // s_new = left @ right (M x B, K = M)
    gemm_f32_wmma<0, 0><<<dim3(Bc / GBN, M / GBM), 256, 0, stream>>>(
        G, right, s_new, nullptr, M, Bc, M, M, Bc, Bc, 0.0f, nullptr, done);
    k_update<<<ndiff, 256, 0, stream>>>(s_out, s_new, u, v, diff_p, nel, done);
    // err = sum((H @ s_new - x)^2)  (N x B, K = M) -> per-block partials
    gemm_f32_wmma<0, 1><<<dim3(Bc / GBN, N / GBM), 256, 0, stream>>>(
        H, s_new, nullptr, x, N, Bc, M, M, Bc, Bc, 0.0f, err_p, done);
    k_flag<<<1, 256, 0, stream>>>(diff_p, ndiff, err_p, nerr, done, errors, it);
  }
}